// AudioVisualBackbone_78340203479485
// MI455X (gfx1250) — compile-verified
//
#include <hip/hip_runtime.h>
#include <math.h>

typedef float v2f __attribute__((ext_vector_type(2)));
typedef float v8f __attribute__((ext_vector_type(8)));

#define B_  64
#define L_  16
#define S_  49
#define VD_ 1028
#define AD_ 1024
#define D_  512
#define H_  256
#define C_  4
#define EPS_ 1e-8f

// ---------------- LDS layout for frame kernel (floats) ----------------
#define FV_STRIDE 516
#define OFF_FV    0
#define SZ_FV     (64 * FV_STRIDE)       // 33024
#define OFF_SA    (OFF_FV + SZ_FV)       // A stage: 64 x 17
#define SZ_SA     (64 * 17)
#define OFF_SB    (OFF_SA + SZ_SA)       // B stage: 512 x 17
#define SZ_SB     (512 * 17)
#define OFF_RAW   (OFF_SB + SZ_SB)       // raw[64], w[64], colsum[512], sws[512] contiguous
#define SMEM_FLOATS (OFF_RAW + 64 + 64 + 512 + 512)

// =====================================================================
// Kernel 1: audio path. f_a = audio@Wap^T + bap ; score_a = tanh(f_a@W_audio^T)
// =====================================================================
__global__ __launch_bounds__(256) void k_audio(
    const float* __restrict__ audio, const float* __restrict__ Wap,
    const float* __restrict__ bap,   const float* __restrict__ W_audio,
    float* __restrict__ fa_ws, float* __restrict__ sca_ws)
{
  __shared__ float a_lds[AD_];
  __shared__ float fa_lds[D_];
  const int b = blockIdx.x, t = threadIdx.x;
  for (int i = t; i < AD_; i += 256) a_lds[i] = audio[(size_t)b * AD_ + i];
  __syncthreads();
  for (int d = t; d < D_; d += 256) {
    const float* wr = Wap + (size_t)d * AD_;
    float s = bap[d];
    for (int v = 0; v < AD_; ++v) s += a_lds[v] * wr[v];
    fa_lds[d] = s;
    fa_ws[(size_t)b * D_ + d] = s;
  }
  __syncthreads();
  for (int e = t; e < D_; e += 256) {
    const float* wr = W_audio + (size_t)e * D_;
    float s = 0.f;
    for (int d = 0; d < D_; ++d) s += fa_lds[d] * wr[d];
    sca_ws[(size_t)b * D_ + e] = tanhf(s);
  }
}

// =====================================================================
// Kernel 2: fused per-frame pipeline with fp32 WMMA.
//   GEMM1: f_v[64,512] = visual_frame[64,1028] x Wvp^T  (+bvp, pad rows zeroed)
//   GEMM2: score_v[64,512] = tanh(f_v x W_visual^T)     (kept in accumulators)
//   then spatial softmax, sws/enh reductions, avg accumulation.
// Block: 256 threads = 8 wave32, waves tiled 4 (M) x 2 (N halves of 256).
// =====================================================================
__global__ __launch_bounds__(256) void k_frame(
    const float* __restrict__ visual, const float* __restrict__ Wvp,
    const float* __restrict__ bvp,    const float* __restrict__ W_visual,
    const float* __restrict__ sca_ws,
    float* __restrict__ enh_ws, float* __restrict__ traw_ws,
    float* __restrict__ avg_ws)
{
  extern __shared__ float sm[];
  float* fv  = sm + OFF_FV;
  float* stA = sm + OFF_SA;
  float* stB = sm + OFF_SB;
  float* raw = sm + OFF_RAW;        // [64]
  float* wsp = raw + 64;            // [64]
  float* cs  = wsp + 64;            // [512] column sums of score_v
  float* sws = cs + 512;            // [512] spatially weighted score

  const int t    = threadIdx.x;
  const int wave = t >> 5, lane = t & 31;
  const int mt   = wave & 3;        // M tile (rows mt*16..+15)
  const int nh   = wave >> 2;       // N half (cols nh*256..+255)
  const int hi   = lane >> 4;       // lane half
  const int ln   = lane & 15;
  const int frame = blockIdx.x;     // b*L + l
  const int b     = frame / L_;
  const float* vis = visual + (size_t)frame * S_ * VD_;

  for (int i = t; i < 64 + 64 + 512 + 512; i += 256) raw[i] = 0.f;

  v8f acc[16] = {};
  __syncthreads();

  // ---------------- GEMM1: K = 1028 in stages of 16 (last stage 4) ----------------
  for (int kb = 0; kb < VD_; kb += 16) {
    const int kt = (VD_ - kb) >= 16 ? 16 : (VD_ - kb);
    // stage A: 64 x kt (zero pad rows >= 49)
    for (int i = t; i < 64 * kt; i += 256) {
      const int r = i / kt, c = i - r * kt;
      stA[r * 17 + c] = (r < S_) ? vis[(size_t)r * VD_ + kb + c] : 0.f;
    }
    // stage B: Wvp tile 512 x kt, float4 loads (rows are 16B aligned: 1028*4 = 257*16)
    if (kt == 16) {
      if (kb + 16 < VD_) __builtin_prefetch(Wvp + (size_t)(t * 2) * VD_ + kb + 16, 0, 0);
      for (int q = t; q < 512 * 4; q += 256) {
        const int n = q >> 2, c4 = (q & 3) << 2;
        const float4 v4 = *(const float4*)(Wvp + (size_t)n * VD_ + kb + c4);
        stB[n * 17 + c4 + 0] = v4.x; stB[n * 17 + c4 + 1] = v4.y;
        stB[n * 17 + c4 + 2] = v4.z; stB[n * 17 + c4 + 3] = v4.w;
      }
    } else {
      for (int n = t; n < 512; n += 256) {
        const float4 v4 = *(const float4*)(Wvp + (size_t)n * VD_ + kb);
        stB[n * 17 + 0] = v4.x; stB[n * 17 + 1] = v4.y;
        stB[n * 17 + 2] = v4.z; stB[n * 17 + 3] = v4.w;
      }
    }
    __syncthreads();
    for (int kk = 0; kk < kt; kk += 4) {
      v2f a;
      a.x = stA[(mt * 16 + ln) * 17 + kk + 2 * hi + 0];
      a.y = stA[(mt * 16 + ln) * 17 + kk + 2 * hi + 1];
      #pragma unroll
      for (int n = 0; n < 16; ++n) {
        const int col = nh * 256 + n * 16 + ln;
        v2f bb;
        bb.x = stB[col * 17 + kk + 2 * hi + 0];
        bb.y = stB[col * 17 + kk + 2 * hi + 1];
        acc[n] = __builtin_amdgcn_wmma_f32_16x16x4_f32(
            false, a, false, bb, (short)0, acc[n], false, false);
      }
    }
    __syncthreads();
  }

  // write f_v (+bias; force padded rows to exactly 0 so tanh path stays clean)
  #pragma unroll
  for (int n = 0; n < 16; ++n) {
    const int col = nh * 256 + n * 16 + ln;
    const float bias = bvp[col];
    #pragma unroll
    for (int r = 0; r < 8; ++r) {
      const int row = mt * 16 + r + 8 * hi;
      fv[row * FV_STRIDE + col] = (row < S_) ? (acc[n][r] + bias) : 0.f;
    }
  }
  __syncthreads();

  // ---------------- GEMM2: score_v = f_v x W_visual^T, K = 512 ----------------
  #pragma unroll
  for (int n = 0; n < 16; ++n)
    #pragma unroll
    for (int r = 0; r < 8; ++r) acc[n][r] = 0.f;

  for (int kb = 0; kb < D_; kb += 16) {
    for (int q = t; q < 512 * 4; q += 256) {
      const int n = q >> 2, c4 = (q & 3) << 2;
      const float4 v4 = *(const float4*)(W_visual + (size_t)n * D_ + kb + c4);
      stB[n * 17 + c4 + 0] = v4.x; stB[n * 17 + c4 + 1] = v4.y;
      stB[n * 17 + c4 + 2] = v4.z; stB[n * 17 + c4 + 3] = v4.w;
    }
    __syncthreads();
    #pragma unroll
    for (int kk = 0; kk < 16; kk += 4) {
      v2f a;
      a.x = fv[(mt * 16 + ln) * FV_STRIDE + kb + kk + 2 * hi + 0];
      a.y = fv[(mt * 16 + ln) * FV_STRIDE + kb + kk + 2 * hi + 1];
      #pragma unroll
      for (int n = 0; n < 16; ++n) {
        const int col = nh * 256 + n * 16 + ln;
        v2f bb;
        bb.x = stB[col * 17 + kk + 2 * hi + 0];
        bb.y = stB[col * 17 + kk + 2 * hi + 1];
        acc[n] = __builtin_amdgcn_wmma_f32_16x16x4_f32(
            false, a, false, bb, (short)0, acc[n], false, false);
      }
    }
    __syncthreads();
  }

  // tanh -> score_v in regs; accumulate raw[s] and column sums
  float rawp[8];
  #pragma unroll
  for (int r = 0; r < 8; ++r) rawp[r] = 0.f;
  #pragma unroll
  for (int n = 0; n < 16; ++n) {
    const int col = nh * 256 + n * 16 + ln;
    const float sa = sca_ws[(size_t)b * D_ + col];
    float csum = 0.f;
    #pragma unroll
    for (int r = 0; r < 8; ++r) {
      const float v = tanhf(acc[n][r]);
      acc[n][r] = v;                 // keep score_v resident in accumulators
      rawp[r] += v * sa;
      csum += v;                     // padded rows contribute tanh(0)=0
    }
    atomicAdd(&cs[col], csum);
  }
  #pragma unroll
  for (int r = 0; r < 8; ++r) {
    const int row = mt * 16 + r + 8 * hi;
    if (row < S_) atomicAdd(&raw[row], rawp[r]);
  }
  __syncthreads();

  // spatial softmax over s = 0..48
  if (t == 0) {
    float m = raw[0];
    for (int s = 1; s < S_; ++s) m = fmaxf(m, raw[s]);
    float sum = 0.f;
    for (int s = 0; s < S_; ++s) { const float e = __expf(raw[s] - m); wsp[s] = e; sum += e; }
    const float inv = 1.f / sum;
    for (int s = 0; s < S_; ++s) wsp[s] *= inv;
    for (int s = S_; s < 64; ++s) wsp[s] = 0.f;
  }
  __syncthreads();

  // sws[d] = sum_s w[s] * score_v[s,d]
  #pragma unroll
  for (int n = 0; n < 16; ++n) {
    const int col = nh * 256 + n * 16 + ln;
    float sacc = 0.f;
    #pragma unroll
    for (int r = 0; r < 8; ++r) {
      const int row = mt * 16 + r + 8 * hi;
      sacc += wsp[row] * acc[n][r];  // wsp[row>=49] == 0
    }
    atomicAdd(&sws[col], sacc);
  }
  __syncthreads();

  // enh[d] = sum_s w[s]*f_v[s,d]; avg accumulation; t_raw = sum_d sws[d]
  for (int d = t; d < D_; d += 256) {
    float e = 0.f;
    for (int s = 0; s < S_; ++s) e += wsp[s] * fv[s * FV_STRIDE + d];
    enh_ws[(size_t)frame * D_ + d] = e;
    atomicAdd(&avg_ws[(size_t)b * D_ + d], cs[d]);
  }
  if (t == 0) {
    float tr = 0.f;
    for (int d = 0; d < D_; ++d) tr += sws[d];
    traw_ws[frame] = tr;
  }
}

// =====================================================================
// Kernel 3: per-batch finalize: temporal softmax, audio attention, cosine
// loss pieces, gated fusion + classifier.
// =====================================================================
__global__ __launch_bounds__(256) void k_finalize(
    const float* __restrict__ fa_ws,  const float* __restrict__ sca_ws,
    const float* __restrict__ enh_ws, const float* __restrict__ traw_ws,
    const float* __restrict__ avg_ws,
    const float* __restrict__ U_visual, const float* __restrict__ U_audio,
    const float* __restrict__ Wc1, const float* __restrict__ bc1,
    const float* __restrict__ Wc2, const float* __restrict__ bc2,
    float* __restrict__ out, float* __restrict__ loss_ws)
{
  __shared__ float wt[L_];
  __shared__ float fpv[D_], fpa[D_], fused[D_], hid[H_];
  __shared__ float red[256];
  const int b = blockIdx.x, t = threadIdx.x;

  if (t == 0) {
    float m = traw_ws[b * L_];
    for (int l = 1; l < L_; ++l) m = fmaxf(m, traw_ws[b * L_ + l]);
    float s = 0.f;
    for (int l = 0; l < L_; ++l) { const float e = __expf(traw_ws[b * L_ + l] - m); wt[l] = e; s += e; }
    const float inv = 1.f / s;
    for (int l = 0; l < L_; ++l) wt[l] *= inv;
  }
  __syncthreads();

  for (int d = t; d < D_; d += 256) {
    float s = 0.f;
    for (int l = 0; l < L_; ++l) s += wt[l] * enh_ws[(size_t)(b * L_ + l) * D_ + d];
    fpv[d] = s;
  }
  __syncthreads();

  // audio attention softmax over D (each thread owns d = t and d = t+256)
  const float g0 = (avg_ws[(size_t)b * D_ + t]       * (1.f / 784.f)) * sca_ws[(size_t)b * D_ + t];
  const float g1 = (avg_ws[(size_t)b * D_ + t + 256] * (1.f / 784.f)) * sca_ws[(size_t)b * D_ + t + 256];
  red[t] = fmaxf(g0, g1); __syncthreads();
  for (int off = 128; off > 0; off >>= 1) { if (t < off) red[t] = fmaxf(red[t], red[t + off]); __syncthreads(); }
  const float gmax = red[0]; __syncthreads();
  const float e0 = __expf(g0 - gmax), e1 = __expf(g1 - gmax);
  red[t] = e0 + e1; __syncthreads();
  for (int off = 128; off > 0; off >>= 1) { if (t < off) red[t] += red[t + off]; __syncthreads(); }
  const float esum = red[0]; __syncthreads();
  fpa[t]       = fa_ws[(size_t)b * D_ + t]       * (e0 / esum);
  fpa[t + 256] = fa_ws[(size_t)b * D_ + t + 256] * (e1 / esum);
  __syncthreads();

  // cosine similarity reductions
  float p = fpv[t] * fpa[t] + fpv[t + 256] * fpa[t + 256];
  red[t] = p; __syncthreads();
  for (int off = 128; off > 0; off >>= 1) { if (t < off) red[t] += red[t + off]; __syncthreads(); }
  const float dot = red[0]; __syncthreads();
  p = fpv[t] * fpv[t] + fpv[t + 256] * fpv[t + 256];
  red[t] = p; __syncthreads();
  for (int off = 128; off > 0; off >>= 1) { if (t < off) red[t] += red[t + off]; __syncthreads(); }
  const float nv2 = red[0]; __syncthreads();
  p = fpa[t] * fpa[t] + fpa[t + 256] * fpa[t + 256];
  red[t] = p; __syncthreads();
  for (int off = 128; off > 0; off >>= 1) { if (t < off) red[t] += red[t + off]; __syncthreads(); }
  const float na2 = red[0]; __syncthreads();
  if (t == 0) {
    const float nv = fmaxf(sqrtf(nv2), EPS_);
    const float na = fmaxf(sqrtf(na2), EPS_);
    loss_ws[b] = 1.f - dot / (nv * na);
  }

  // gated fusion
  for (int e = t; e < D_; e += 256) {
    const float* uv = U_visual + (size_t)e * D_;
    const float* ua = U_audio  + (size_t)e * D_;
    float s1 = 0.f, s2 = 0.f;
    for (int d = 0; d < D_; ++d) { s1 += fpv[d] * uv[d]; s2 += fpa[d] * ua[d]; }
    fused[e] = 1.f / (1.f + __expf(-s1)) + 1.f / (1.f + __expf(-s2));
  }
  __syncthreads();
  for (int h = t; h < H_; h += 256) {
    const float* w = Wc1 + (size_t)h * D_;
    float s = bc1[h];
    for (int e = 0; e < D_; ++e) s += fused[e] * w[e];
    hid[h] = fmaxf(s, 0.f);
  }
  __syncthreads();
  if (t < C_) {
    const float* w = Wc2 + (size_t)t * H_;
    float s = bc2[t];
    for (int h = 0; h < H_; ++h) s += hid[h] * w[h];
    out[b * C_ + t] = s;
  }
}

// =====================================================================
// Kernel 4: mean of per-batch losses -> out[256]
// =====================================================================
__global__ void k_loss(const float* __restrict__ loss_ws, float* __restrict__ out) {
  __shared__ float r[64];
  const int t = threadIdx.x;
  r[t] = loss_ws[t];
  __syncthreads();
  for (int off = 32; off > 0; off >>= 1) { if (t < off) r[t] += r[t + off]; __syncthreads(); }
  if (t == 0) out[B_ * C_] = r[0] / (float)B_;
}

// =====================================================================
extern "C" void kernel_launch(void* const* d_in, const int* in_sizes, int n_in,
                              void* d_out, int out_size, void* d_ws, size_t ws_size,
                              hipStream_t stream) {
  const float* visual   = (const float*)d_in[0];
  const float* audio    = (const float*)d_in[1];
  const float* Wvp      = (const float*)d_in[2];
  const float* bvp      = (const float*)d_in[3];
  const float* Wap      = (const float*)d_in[4];
  const float* bap      = (const float*)d_in[5];
  const float* W_audio  = (const float*)d_in[6];
  const float* W_visual = (const float*)d_in[7];
  const float* U_visual = (const float*)d_in[8];
  const float* U_audio  = (const float*)d_in[9];
  const float* Wc1      = (const float*)d_in[10];
  const float* bc1      = (const float*)d_in[11];
  const float* Wc2      = (const float*)d_in[12];
  const float* bc2      = (const float*)d_in[13];
  float* out = (float*)d_out;

  float* ws      = (float*)d_ws;
  float* fa_ws   = ws;                       // [64,512]
  float* sca_ws  = fa_ws   + B_ * D_;        // [64,512]
  float* enh_ws  = sca_ws  + B_ * D_;        // [64,16,512]
  float* traw_ws = enh_ws  + B_ * L_ * D_;   // [64,16]
  float* avg_ws  = traw_ws + B_ * L_;        // [64,512]
  float* loss_ws = avg_ws  + B_ * D_;        // [64]

  const size_t smem = SMEM_FLOATS * sizeof(float);
  hipFuncSetAttribute((const void*)k_frame,
                      hipFuncAttributeMaxDynamicSharedMemorySize, (int)smem);

  hipMemsetAsync(avg_ws, 0, B_ * D_ * sizeof(float), stream);

  hipLaunchKernelGGL(k_audio, dim3(B_), dim3(256), 0, stream,
                     audio, Wap, bap, W_audio, fa_ws, sca_ws);
  hipLaunchKernelGGL(k_frame, dim3(B_ * L_), dim3(256), smem, stream,
                     visual, Wvp, bvp, W_visual, sca_ws, enh_ws, traw_ws, avg_ws);
  hipLaunchKernelGGL(k_finalize, dim3(B_), dim3(256), 0, stream,
                     fa_ws, sca_ws, enh_ws, traw_ws, avg_ws,
                     U_visual, U_audio, Wc1, bc1, Wc2, bc2, out, loss_ws);
  hipLaunchKernelGGL(k_loss, dim3(1), dim3(64), 0, stream, loss_ws, out);
}